// GINEEncoderAdapter_84799834292893
// MI455X (gfx1250) — compile-verified
//
#include <hip/hip_runtime.h>
#include <hip/hip_bf16.h>
#include <cstdint>

// Problem constants (fixed by the reference).
constexpr int D          = 256;          // embedding width (floats)
constexpr int NUM_GRAPHS = 1024;
constexpr int MAX_NODES  = 512;

// Tiling.
constexpr int ROWS_PER_BLOCK = 64;                 // 64 rows * 1KB = 64KB LDS
constexpr int THREADS        = 256;                // 8 waves (wave32)
constexpr int CHUNKS         = D / 4;              // 64 float4 chunks per row
constexpr int ROWS_PER_ITER  = THREADS / CHUNKS;   // 4 rows per iteration
constexpr int NUM_ITERS      = ROWS_PER_BLOCK / ROWS_PER_ITER; // 16

// Native clang vector type (legal for __builtin_nontemporal_store).
typedef __attribute__((ext_vector_type(4))) float v4f;

// ---------------------------------------------------------------------------
// Kernel 1: starts/counts via binary search on the *sorted* batch_idx.
// O(G log N), no atomics, deterministic.
// ---------------------------------------------------------------------------
__global__ __launch_bounds__(256)
void gine_offsets_kernel(const int* __restrict__ bidx, int n,
                         int* __restrict__ starts, int* __restrict__ counts) {
    int g = blockIdx.x * blockDim.x + threadIdx.x;
    if (g >= NUM_GRAPHS) return;
    // lower_bound(g)
    int lo = 0, hi = n;
    while (lo < hi) { int m = (lo + hi) >> 1; if (bidx[m] < g) lo = m + 1; else hi = m; }
    int s = lo;
    // lower_bound(g+1)
    lo = s; hi = n;
    while (lo < hi) { int m = (lo + hi) >> 1; if (bidx[m] < g + 1) lo = m + 1; else hi = m; }
    starts[g] = s;
    counts[g] = lo - s;
}

// Flat shared-memory pointer -> 32-bit LDS byte offset (addr[31:0] is the
// in-allocation LDS offset; hardware adds LDS_BASE for async ops).
__device__ __forceinline__ unsigned lds_off32(const void* p) {
    return (unsigned)(uintptr_t)p;
}

// ---------------------------------------------------------------------------
// Kernel 2: streamed block copy through the async global<->LDS engine.
// All traffic is stream-once -> non-temporal hints keep it out of L2's
// RT working set.
//   valid rows   : global_load_async_to_lds_b128 (NT) -> s_wait_asynccnt ->
//                  global_store_async_from_lds_b128 (NT)
//   invalid rows : direct NT b128 zero stores (no read traffic)
//   mask         : fused (first 64 lanes of each block)
// ---------------------------------------------------------------------------
__global__ __launch_bounds__(THREADS)
void gine_scatter_kernel(const float* __restrict__ node_emb,
                         const int* __restrict__ starts,
                         const int* __restrict__ counts,
                         float* __restrict__ enc_states,
                         float* __restrict__ enc_mask) {
    __shared__ float smem[ROWS_PER_BLOCK * D];   // 64 KB staging buffer

    const int g     = blockIdx.x;                // graph id
    const int slab  = blockIdx.y;                // which group of 64 slots
    const int slot0 = slab * ROWS_PER_BLOCK;
    const int start = starts[g];
    const int count = counts[g];

    const int tid   = threadIdx.x;
    const int chunk = tid & (CHUNKS - 1);        // float4 index within row
    const int rsub  = tid >> 6;                  // 0..3 (row within iteration)

    // -------- phase 1: async gather of valid rows into LDS ----------------
    #pragma unroll
    for (int it = 0; it < NUM_ITERS; ++it) {
        const int rlocal = it * ROWS_PER_ITER + rsub;
        const int slot   = slot0 + rlocal;
        if (slot < count) {
            const float* src = node_emb + (size_t)(start + slot) * D + chunk * 4;
            unsigned loff = lds_off32(&smem[rlocal * D + chunk * 4]);
            asm volatile("global_load_async_to_lds_b128 %0, %1, off th:TH_LOAD_NT"
                         :: "v"(loff), "v"(src) : "memory");
        }
    }

    // wait for all async loads before reading LDS via async stores
    asm volatile("s_wait_asynccnt 0" ::: "memory");

    // -------- phase 2: async scatter to contiguous destinations -----------
    #pragma unroll
    for (int it = 0; it < NUM_ITERS; ++it) {
        const int rlocal = it * ROWS_PER_ITER + rsub;
        const int slot   = slot0 + rlocal;
        float* dst = enc_states + ((size_t)g * MAX_NODES + slot) * D + chunk * 4;
        if (slot < count) {
            unsigned loff = lds_off32(&smem[rlocal * D + chunk * 4]);
            asm volatile("global_store_async_from_lds_b128 %0, %1, off th:TH_STORE_NT"
                         :: "v"(dst), "v"(loff) : "memory");
        } else {
            v4f z = (v4f)(0.0f);
            __builtin_nontemporal_store(z, reinterpret_cast<v4f*>(dst));
        }
    }
    // trailing async stores are covered by S_ENDPGM's implicit wait-idle

    // -------- fused mask write --------------------------------------------
    if (tid < ROWS_PER_BLOCK) {
        const int slot = slot0 + tid;
        float m = (slot < count) ? 1.0f : 0.0f;
        __builtin_nontemporal_store(m, &enc_mask[(size_t)g * MAX_NODES + slot]);
    }
}

// ---------------------------------------------------------------------------
extern "C" void kernel_launch(void* const* d_in, const int* in_sizes, int n_in,
                              void* d_out, int out_size, void* d_ws, size_t ws_size,
                              hipStream_t stream) {
    const float* node_emb  = (const float*)d_in[0];
    const int*   batch_idx = (const int*)d_in[1];
    const int n = in_sizes[1];                      // TOTAL_NODES

    int* starts = (int*)d_ws;                       // [NUM_GRAPHS]
    int* counts = starts + NUM_GRAPHS;              // [NUM_GRAPHS]

    gine_offsets_kernel<<<dim3((NUM_GRAPHS + 255) / 256), dim3(256), 0, stream>>>(
        batch_idx, n, starts, counts);

    float* enc_states = (float*)d_out;
    float* enc_mask   = enc_states + (size_t)NUM_GRAPHS * MAX_NODES * D;

    gine_scatter_kernel<<<dim3(NUM_GRAPHS, MAX_NODES / ROWS_PER_BLOCK),
                          dim3(THREADS), 0, stream>>>(
        node_emb, starts, counts, enc_states, enc_mask);
}